// HomogeneousFiveTypeGINEncoder_87686052315190
// MI455X (gfx1250) — compile-verified
//
#include <hip/hip_runtime.h>

typedef float v2f __attribute__((ext_vector_type(2)));
typedef float v8f __attribute__((ext_vector_type(8)));

#define N_TOTAL 50000
#define NPT     10000
#define ICH     128
#define EMB     8
#define C0      136   // layer-0 feature width (128 + 8 type emb)
#define HID     128

// ---------------------------------------------------------------- zero
__global__ void zero_kernel(float* __restrict__ p, int n) {
    int i = blockIdx.x * blockDim.x + threadIdx.x;
    if (i < n) p[i] = 0.0f;
}

// ---------------------------------------------------------------- h0 = concat(x_all, type_emb[type])
__global__ void build_h0(const float* __restrict__ xp, const float* __restrict__ xpl,
                         const float* __restrict__ xg, const float* __restrict__ xsg,
                         const float* __restrict__ xst, const float* __restrict__ temb,
                         float* __restrict__ h0) {
    int i = blockIdx.x * blockDim.x + threadIdx.x;
    if (i >= N_TOTAL * C0) return;
    int n = i / C0;
    int c = i - n * C0;
    int t = n / NPT;
    int r = n - t * NPT;
    float v;
    if (c < ICH) {
        const float* x = (t == 0) ? xp : (t == 1) ? xpl : (t == 2) ? xg : (t == 3) ? xsg : xst;
        v = x[r * ICH + c];
    } else {
        v = temb[t * EMB + (c - ICH)];
    }
    h0[i] = v;
}

// ---------------------------------------------------------------- edge scatter: agg[dst] += h[src]
// One thread = one float4 chunk of one edge. Consecutive tids walk consecutive
// chunks of the same edge -> coalesced float4 gathers + coalesced f32 atomics.
// h (<=27MB) is L2-resident, so this is pure L2 gather/atomic bandwidth.
__global__ void scatter_add(const float* __restrict__ h, float* __restrict__ agg,
                            const int* __restrict__ src, const int* __restrict__ dst,
                            int E, int C) {
    int c4 = C >> 2;
    unsigned tid = blockIdx.x * blockDim.x + threadIdx.x;
    unsigned total = (unsigned)E * (unsigned)c4;
    if (tid >= total) return;
    unsigned e  = tid / (unsigned)c4;
    unsigned ch = tid - e * (unsigned)c4;
    int s = src[e];
    int d = dst[e];
    float4 v = *((const float4*)(h + (size_t)s * C) + ch);
    float* ap = agg + (size_t)d * C + (size_t)ch * 4u;
    atomicAdd(ap + 0, v.x);
    atomicAdd(ap + 1, v.y);
    atomicAdd(ap + 2, v.z);
    atomicAdd(ap + 3, v.w);
}

// ---------------------------------------------------------------- fused (X[+Xadd]) @ W + bias, ReLU
// Block = 256 threads = 8 waves. Each block: one 16-row tile, N = 128 outputs.
// z-tile staged in LDS; each wave owns a 16-wide column strip and runs a
// K-loop of V_WMMA_F32_16X16X4_F32 (exact fp32, matches reference numerics).
__global__ __launch_bounds__(256) void gin_mlp_gemm(
        const float* __restrict__ X, const float* __restrict__ Xadd,
        const float* __restrict__ W, const float* __restrict__ bias,
        float* __restrict__ Y, int K) {
    __shared__ float zs[16 * C0];            // max K = 136 -> 8.5 KB
    const int tid  = threadIdx.x;
    const size_t rowbase = (size_t)blockIdx.x * 16;

    // stage z = X (+ Xadd) for this 16-row tile
    for (int i = tid; i < 16 * K; i += 256) {
        int r = i / K;
        int c = i - r * K;
        size_t off = (rowbase + r) * (size_t)K + c;
        float v = X[off];
        if (Xadd) v += Xadd[off];
        zs[r * K + c] = v;
    }
    __syncthreads();

    const int wave = tid >> 5;
    const int lane = tid & 31;
    const int half = lane >> 4;       // 0: K/M-low group, 1: high group
    const int m    = lane & 15;       // A: row M,  B/C/D: column N
    const int n0   = wave << 4;       // this wave's 16-column strip

    // accumulator initialized with bias (same column for all 8 elems of a lane)
    const float bv = bias[n0 + m];
    v8f acc = {bv, bv, bv, bv, bv, bv, bv, bv};

    const float* wp = W + n0 + m;     // W is [K,128] row-major
    for (int k = 0; k < K; k += 4) {
        v2f a, b;
        // A 16x4 f32: lane m = M; VGPR0 -> K = k+2*half, VGPR1 -> K = k+2*half+1
        a.x = zs[m * K + k + 2 * half];
        a.y = zs[m * K + k + 2 * half + 1];
        // B 4x16 f32: lane m = N; same K striping as A
        b.x = wp[(k + 2 * half) * HID];
        b.y = wp[(k + 2 * half + 1) * HID];
        acc = __builtin_amdgcn_wmma_f32_16x16x4_f32(
                  false, a, false, b, (short)0, acc, false, false);
    }

    // C/D layout: VGPR r, lane -> row = r + 8*half, col = n0 + m ; fused ReLU
    float* yp = Y + (rowbase + (size_t)(half * 8)) * HID + n0 + m;
#pragma unroll
    for (int r = 0; r < 8; ++r) {
        float v = acc[r];
        yp[r * HID] = v > 0.0f ? v : 0.0f;
    }
}

// ---------------------------------------------------------------- out = h[:10000]
__global__ void copy_out(const float* __restrict__ h, float* __restrict__ out) {
    int i = blockIdx.x * blockDim.x + threadIdx.x;
    if (i < NPT * HID) out[i] = h[i];
}

// ----------------------------------------------------------------
extern "C" void kernel_launch(void* const* d_in, const int* in_sizes, int n_in,
                              void* d_out, int out_size, void* d_ws, size_t ws_size,
                              hipStream_t stream) {
    const float* xp   = (const float*)d_in[0];
    const float* xpl  = (const float*)d_in[1];
    const float* xg   = (const float*)d_in[2];
    const float* xsg  = (const float*)d_in[3];
    const float* xst  = (const float*)d_in[4];
    const int*   ei   = (const int*)d_in[5];
    const float* temb = (const float*)d_in[6];
    const float* W0s[3] = {(const float*)d_in[7],  (const float*)d_in[11], (const float*)d_in[15]};
    const float* B0s[3] = {(const float*)d_in[8],  (const float*)d_in[12], (const float*)d_in[16]};
    const float* W1s[3] = {(const float*)d_in[9],  (const float*)d_in[13], (const float*)d_in[17]};
    const float* B1s[3] = {(const float*)d_in[10], (const float*)d_in[14], (const float*)d_in[18]};

    const int E = in_sizes[5] / 2;
    const int* src = ei;
    const int* dst = ei + E;

    // workspace arenas (floats): A = h (N x 136 max), B = agg, C = MLP hidden
    float* A  = (float*)d_ws;
    float* B  = A + (size_t)N_TOTAL * C0;
    float* Cb = B + (size_t)N_TOTAL * C0;

    // h0 = [x_all | type_emb]
    {
        int n = N_TOTAL * C0;
        build_h0<<<(n + 255) / 256, 256, 0, stream>>>(xp, xpl, xg, xsg, xst, temb, A);
    }

    for (int l = 0; l < 3; ++l) {
        const int C = (l == 0) ? C0 : HID;

        // agg = 0
        int nz = N_TOTAL * C;
        zero_kernel<<<(nz + 255) / 256, 256, 0, stream>>>(B, nz);

        // agg[dst] += h[src]
        unsigned total = (unsigned)E * (unsigned)(C / 4);
        scatter_add<<<(total + 255u) / 256u, 256, 0, stream>>>(A, B, src, dst, E, C);

        // hidden = relu((h + agg) @ W0 + b0)
        gin_mlp_gemm<<<N_TOTAL / 16, 256, 0, stream>>>(A, B, W0s[l], B0s[l], Cb, C);
        // h = relu(hidden @ W1 + b1)   (writes back into arena A, stride 128)
        gin_mlp_gemm<<<N_TOTAL / 16, 256, 0, stream>>>(Cb, nullptr, W1s[l], B1s[l], A, HID);
    }

    copy_out<<<(NPT * HID + 255) / 256, 256, 0, stream>>>(A, (float*)d_out);
}